// SoftNCutsLoss_v1_62672162784079
// MI455X (gfx1250) — compile-verified
//
#include <hip/hip_runtime.h>

typedef float v2f __attribute__((ext_vector_type(2)));
typedef float v4f __attribute__((ext_vector_type(4)));
typedef float v8f __attribute__((ext_vector_type(8)));

#define N_VOX 4096
#define KCLS  6
#define PT_STRIDE 4100   // padded: bank = (4*t) % 64 -> conflict-free b128 reads

// exp(-d^2/9) = exp2( -( (f_i - f_j) * KSCALE )^2 ),  KSCALE = sqrt(log2 e)/3
#define KSCALE 0.40037414f

// -------------------------------------------------------------------------
// One pass over dist_weight (67 MB @ 23.3 TB/s ~ 3 us floor). Each wave owns
// 16 rows; per 32-column chunk: 4x global_load_b128 (dist stream, per-lane
// contiguous), 4x ds_load_b128 (f*k, broadcast), 4x ds_load_b128 (P^T col),
// then 8x v_wmma_f32_16x16x4_f32. Chunks are double-buffered so loads for
// chunk n+1 overlap the WMMA/exp work of chunk n. exp uses the raw
// v_exp_f32 (argument is in [-0.161, 0] -> no denormal guard needed).
// LDS: P^T extended [8][4100] (rows 6=ones,7=zeros) + f*k [4096] = 147.6 KB.
// -------------------------------------------------------------------------
__global__ void __launch_bounds__(256)
softncuts_main(const float* __restrict__ patch,
               const float* __restrict__ prob,
               const float* __restrict__ dist,
               float* __restrict__ ws_num,
               float* __restrict__ ws_den)
{
    extern __shared__ float lds[];
    float* ldsPT = lds;                       // [8][PT_STRIDE]
    float* ldsF  = lds + 8 * PT_STRIDE;       // [N_VOX], pre-scaled by KSCALE

    const int tid = threadIdx.x;

    // Stage P^T (+ ones / zeros rows) and f*k into LDS once per block.
    for (int idx = tid; idx < N_VOX; idx += 256) {
        const float* pr = prob + idx * KCLS;
#pragma unroll
        for (int c = 0; c < KCLS; ++c) ldsPT[c * PT_STRIDE + idx] = pr[c];
        ldsPT[6 * PT_STRIDE + idx] = 1.0f;    // rowsum column
        ldsPT[7 * PT_STRIDE + idx] = 0.0f;    // padding columns 7..15
        ldsF[idx] = patch[idx] * KSCALE;
    }
    __syncthreads();

    const int lane = tid & 31;
    const int wave = tid >> 5;
    const int half = lane >> 4;     // 0: K slots {0,1}; 1: K slots {2,3}
    const int t    = lane & 15;     // A row-in-tile; B/C column (class)
    const int m0   = blockIdx.x * 128 + wave * 16;
    const int mRow = m0 + t;

    const float fik = ldsF[mRow];   // f_i * KSCALE
    const float* distRow = dist + (size_t)mRow * N_VOX;

    // Per-lane contiguous 16-float streams for a 32-column chunk.
    const int    baseC = half * 16;
    const float* gRow  = distRow + baseC;
    const float* fRow  = ldsF + baseC;
    const float* bRow  = ldsPT + (size_t)((t < 7) ? t : 7) * PT_STRIDE + baseC;

    v8f acc = {};   // D: rows m0 + v + 8*half, col t; col 6 == rowsum

    auto LOAD = [&](v4f* dw, v4f* fj, v4f* bv, int j0) {
#pragma unroll
        for (int q = 0; q < 4; ++q) {
            dw[q] = *(const v4f*)(gRow + j0 + 4 * q);   // global b128
            fj[q] = *(const v4f*)(fRow + j0 + 4 * q);   // ds b128 (broadcast)
            bv[q] = *(const v4f*)(bRow + j0 + 4 * q);   // ds b128
        }
        __builtin_prefetch(distRow + j0 + 1024, 0, 1);  // 4 KB ahead (speculative)
    };

    auto COMPUTE = [&](const v4f* dw, const v4f* fj, const v4f* bv) {
#pragma unroll
        for (int q = 0; q < 4; ++q) {
#pragma unroll
            for (int h = 0; h < 2; ++h) {
                const float d0 = fik - fj[q][2 * h];
                const float d1 = fik - fj[q][2 * h + 1];
                v2f A, B;
                // raw v_exp_f32: arg in [-0.161, 0], no denorm guard needed
                A.x = __builtin_amdgcn_exp2f(-(d0 * d0)) * dw[q][2 * h];
                A.y = __builtin_amdgcn_exp2f(-(d1 * d1)) * dw[q][2 * h + 1];
                B.x = bv[q][2 * h];
                B.y = bv[q][2 * h + 1];
                acc = __builtin_amdgcn_wmma_f32_16x16x4_f32(
                    false, A, false, B, (short)0, acc, false, false);
            }
        }
    };

    // Double-buffered main loop: 128 chunks of 32 columns.
    v4f dwA[4], fjA[4], bA[4];
    v4f dwB[4], fjB[4], bB[4];
    LOAD(dwA, fjA, bA, 0);
    for (int j0 = 0; j0 < N_VOX; j0 += 64) {
        LOAD(dwB, fjB, bB, j0 + 32);
        COMPUTE(dwA, fjA, bA);
        if (j0 + 64 < N_VOX) LOAD(dwA, fjA, bA, j0 + 64);
        COMPUTE(dwB, fjB, bB);
    }

    // Reduce: num[t] += P[m,t]*WP[m,t]; den[t] += P[m,t]*rowsum[m].
    // rowsum for row group v of this half lives in lane (half*16 | 6).
    const float bScale = (t < 6) ? 1.0f : 0.0f;
    const int   rsLane = (lane & 16) | 6;
    const float* pCol  = ldsPT + (size_t)((t < 7) ? t : 7) * PT_STRIDE;
    float pnum = 0.0f, pden = 0.0f;
#pragma unroll
    for (int v = 0; v < 8; ++v) {
        const float wp = acc[v];
        const float rs = __shfl(wp, rsLane, 32);
        const int   mv = m0 + v + half * 8;
        const float pv = pCol[mv] * bScale;   // 0 for t >= 6
        pnum += pv * wp;
        pden += pv * rs;
    }

    if (t < 6) {   // both halves contribute (disjoint row groups)
        atomicAdd(ws_num + t, pnum);
        atomicAdd(ws_den + t, pden);
    }
}

__global__ void softncuts_zero(float* ws)
{
    if (threadIdx.x < 12) ws[threadIdx.x] = 0.0f;
}

__global__ void softncuts_finalize(const float* __restrict__ ws,
                                   float* __restrict__ out)
{
    if (threadIdx.x == 0) {
        float s = 0.0f;
#pragma unroll
        for (int i = 0; i < KCLS; ++i) s += ws[i] / ws[KCLS + i];
        out[0] = (float)KCLS - s;
    }
}

extern "C" void kernel_launch(void* const* d_in, const int* in_sizes, int n_in,
                              void* d_out, int out_size, void* d_ws, size_t ws_size,
                              hipStream_t stream)
{
    (void)in_sizes; (void)n_in; (void)out_size; (void)ws_size;
    const float* patch = (const float*)d_in[0];   // [4096]
    const float* prob  = (const float*)d_in[1];   // [4096,6]
    const float* dist  = (const float*)d_in[2];   // [4096,4096]
    // d_in[3] = k (==6, compile-time constant here)

    float* ws  = (float*)d_ws;    // ws[0..5]=num, ws[6..11]=den
    float* out = (float*)d_out;

    const size_t ldsBytes = (size_t)(8 * PT_STRIDE + N_VOX) * sizeof(float); // 147584 B

    softncuts_zero<<<1, 32, 0, stream>>>(ws);
    softncuts_main<<<32, 256, ldsBytes, stream>>>(patch, prob, dist, ws, ws + KCLS);
    softncuts_finalize<<<1, 32, 0, stream>>>(ws, out);
}